// GCN_63496796504384
// MI455X (gfx1250) — compile-verified
//
#include <hip/hip_runtime.h>

#define N_NODES 100000
#define N_EDGES 800000
#define F_IN 64
#define F_HID 64
#define F_OUT 16

typedef __attribute__((ext_vector_type(2))) float v2f;
typedef __attribute__((ext_vector_type(8))) float v8f;

// ---------------------------------------------------------------------------
// WMMA wrapper: D = A(16x4,f32) * B(4x16,f32) + C(16x16,f32)
// 8-arg pattern: (neg_a, A, neg_b, B, c_mod, C, reuse_a, reuse_b)
// ---------------------------------------------------------------------------
__device__ __forceinline__ v8f wmma_16x16x4_f32(v2f a, v2f b, v8f c) {
#if defined(__gfx1250__)
  return __builtin_amdgcn_wmma_f32_16x16x4_f32(false, a, false, b, (short)0, c,
                                               false, false);
#else
  return c;
#endif
}

// ---------------------------------------------------------------------------
// Degree / normalization kernels
// ---------------------------------------------------------------------------
__global__ __launch_bounds__(256) void deg_init_kernel(float* deg, int n) {
  int i = blockIdx.x * blockDim.x + threadIdx.x;
  if (i < n) deg[i] = 1.0f;  // self loop contributes 1
}

__global__ __launch_bounds__(256) void deg_accum_kernel(
    const long long* __restrict__ col, float* deg, int e) {
  int i = blockIdx.x * blockDim.x + threadIdx.x;
  if (i < e) atomicAdd(&deg[(int)col[i]], 1.0f);
}

__global__ __launch_bounds__(256) void dinv_kernel(const float* __restrict__ deg,
                                                   float* dinv, int n) {
  int i = blockIdx.x * blockDim.x + threadIdx.x;
  if (i < n) dinv[i] = rsqrtf(deg[i]);  // deg >= 1, no zero case
}

// ---------------------------------------------------------------------------
// WMMA GEMM: Out[n_rows, NO] = A[n_rows, 64] * W[64, NO]
// One wave per 16-row tile; NO/16 column tiles; K=64 in 16 chunks of 4.
// A frag (16x4 f32): lanes 0-15 -> M=lane, VGPR0/1 = K0/K1;
//                    lanes 16-31 -> M=lane-16, VGPR0/1 = K2/K3.
// B frag (4x16 f32): mirrored with N across lanes.
// C/D (16x16 f32): VGPR r, lanes 0-15 -> M=r, N=lane; lanes 16-31 -> M=r+8.
// ---------------------------------------------------------------------------
template <int NO>
__global__ __launch_bounds__(256) void gemm_wmma_kernel(
    const float* __restrict__ A, const float* __restrict__ W,
    float* __restrict__ Out, int n_rows) {
  constexpr int NT = NO / 16;
  const int wave = (int)((blockIdx.x * blockDim.x + threadIdx.x) >> 5);
  const int lane = threadIdx.x & 31;
  const int row0 = wave << 4;
  if (row0 >= n_rows) return;  // wave-uniform exit, EXEC stays all-ones

  const int m  = lane & 15;         // row (A) / col (B,C) within tile
  const int kh = (lane >> 4) << 1;  // K sub-offset: 0 or 2

  v8f acc[NT];
#pragma unroll
  for (int t = 0; t < NT; ++t) acc[t] = (v8f)(0.0f);

  const float* arow = A + (size_t)(row0 + m) * 64 + kh;
#pragma unroll
  for (int kc = 0; kc < 16; ++kc) {
    v2f a;
    a.x = arow[kc * 4 + 0];
    a.y = arow[kc * 4 + 1];
    const float* w0 = W + (size_t)(kc * 4 + kh) * NO + m;
#pragma unroll
    for (int t = 0; t < NT; ++t) {
      v2f b;
      b.x = w0[t * 16];
      b.y = w0[t * 16 + NO];
      acc[t] = wmma_16x16x4_f32(a, b, acc[t]);
    }
  }

  const int mo = (lane >> 4) << 3;  // 0 or 8
#pragma unroll
  for (int t = 0; t < NT; ++t)
#pragma unroll
    for (int r = 0; r < 8; ++r)
      Out[(size_t)(row0 + mo + r) * NO + t * 16 + m] = acc[t][r];
}

// ---------------------------------------------------------------------------
// acc[i][f] = h[i][f] * dinv[i]^2 + bias[f]   (self-loop term + bias)
// ---------------------------------------------------------------------------
template <int NF>
__global__ __launch_bounds__(256) void self_init_kernel(
    const float* __restrict__ h, const float* __restrict__ dinv,
    const float* __restrict__ bias, float* __restrict__ acc, int n) {
  int i = blockIdx.x * blockDim.x + threadIdx.x;
  if (i < n * NF) {
    int node = i / NF;
    int f = i % NF;
    float d = dinv[node];
    acc[i] = h[i] * d * d + bias[f];
  }
}

// ---------------------------------------------------------------------------
// Layer-1 edge scatter: 16 lanes per edge, float4 gather, 4 f32 atomics each.
// ---------------------------------------------------------------------------
__global__ __launch_bounds__(256) void scatter64_kernel(
    const long long* __restrict__ row, const long long* __restrict__ col,
    const float* __restrict__ dinv, const float* __restrict__ h,
    float* __restrict__ acc, int e) {
  int t = blockIdx.x * blockDim.x + threadIdx.x;
  int ed = t >> 4;
  int sub = t & 15;
  if (ed >= e) return;
  int r = (int)row[ed];
  int c = (int)col[ed];
  float norm = dinv[r] * dinv[c];
  const float4 hv = *(const float4*)(h + (size_t)r * 64 + sub * 4);
  float* dst = acc + (size_t)c * 64 + sub * 4;
  atomicAdd(dst + 0, hv.x * norm);
  atomicAdd(dst + 1, hv.y * norm);
  atomicAdd(dst + 2, hv.z * norm);
  atomicAdd(dst + 3, hv.w * norm);
}

__global__ __launch_bounds__(256) void relu_kernel(float* a, int n) {
  int i = blockIdx.x * blockDim.x + threadIdx.x;
  if (i < n) a[i] = fmaxf(a[i], 0.0f);
}

// ---------------------------------------------------------------------------
// Layer-2 edge scatter: 16 lanes per edge, 1 float + 1 atomic each.
// ---------------------------------------------------------------------------
__global__ __launch_bounds__(256) void scatter16_kernel(
    const long long* __restrict__ row, const long long* __restrict__ col,
    const float* __restrict__ dinv, const float* __restrict__ h,
    float* __restrict__ out, int e) {
  int t = blockIdx.x * blockDim.x + threadIdx.x;
  int ed = t >> 4;
  int sub = t & 15;
  if (ed >= e) return;
  int r = (int)row[ed];
  int c = (int)col[ed];
  float norm = dinv[r] * dinv[c];
  float v = h[(size_t)r * 16 + sub] * norm;
  atomicAdd(out + (size_t)c * 16 + sub, v);
}

// ---------------------------------------------------------------------------
// Launch
// ---------------------------------------------------------------------------
extern "C" void kernel_launch(void* const* d_in, const int* in_sizes, int n_in,
                              void* d_out, int out_size, void* d_ws,
                              size_t ws_size, hipStream_t stream) {
  const float* x      = (const float*)d_in[0];            // [N, 64]
  const long long* ei = (const long long*)d_in[1];        // [2, E] int64
  const float* W1     = (const float*)d_in[2];            // [64, 64]
  const float* b1     = (const float*)d_in[3];            // [64]
  const float* W2     = (const float*)d_in[4];            // [64, 16]
  const float* b2     = (const float*)d_in[5];            // [16]
  float* out          = (float*)d_out;                    // [N, 16]

  const long long* row = ei;            // sources
  const long long* col = ei + N_EDGES;  // targets (aggregate here)

  // Workspace layout (floats)
  float* ws   = (float*)d_ws;
  float* deg  = ws;                          // N
  float* dinv = deg + N_NODES;               // N
  float* h1   = dinv + N_NODES;              // N*64
  float* acc1 = h1 + (size_t)N_NODES * 64;   // N*64 (becomes relu'd h2)
  float* h3   = acc1 + (size_t)N_NODES * 64; // N*16

  const int B = 256;
  auto grid = [](long long n, int b) { return (unsigned)((n + b - 1) / b); };

  // 1) degrees + dinv
  deg_init_kernel<<<grid(N_NODES, B), B, 0, stream>>>(deg, N_NODES);
  deg_accum_kernel<<<grid(N_EDGES, B), B, 0, stream>>>(col, deg, N_EDGES);
  dinv_kernel<<<grid(N_NODES, B), B, 0, stream>>>(deg, dinv, N_NODES);

  // 2) h1 = x @ W1   (WMMA f32, one wave per 16 rows -> 6250 waves)
  const long long g1_threads = (long long)(N_NODES / 16) * 32;
  gemm_wmma_kernel<64><<<grid(g1_threads, B), B, 0, stream>>>(x, W1, h1,
                                                              N_NODES);

  // 3) acc1 = h1*dinv^2 + b1, then edge scatter, then ReLU in place
  self_init_kernel<64><<<grid((long long)N_NODES * 64, B), B, 0, stream>>>(
      h1, dinv, b1, acc1, N_NODES);
  scatter64_kernel<<<grid((long long)N_EDGES * 16, B), B, 0, stream>>>(
      row, col, dinv, h1, acc1, N_EDGES);
  relu_kernel<<<grid((long long)N_NODES * 64, B), B, 0, stream>>>(
      acc1, N_NODES * 64);

  // 4) h3 = relu(acc1) @ W2  (WMMA f32, 16 out features)
  gemm_wmma_kernel<16><<<grid(g1_threads, B), B, 0, stream>>>(acc1, W2, h3,
                                                              N_NODES);

  // 5) out = h3*dinv^2 + b2, then edge scatter
  self_init_kernel<16><<<grid((long long)N_NODES * 16, B), B, 0, stream>>>(
      h3, dinv, b2, out, N_NODES);
  scatter16_kernel<<<grid((long long)N_EDGES * 16, B), B, 0, stream>>>(
      row, col, dinv, h3, out, N_EDGES);
}